// GCNRelationPredictor_56332791054339
// MI455X (gfx1250) — compile-verified
//
#include <hip/hip_runtime.h>
#include <hip/hip_bf16.h>

// ---------------------------------------------------------------------------
// 2-layer GCN (PyG GCNConv semantics) for MI455X / gfx1250.
//   - fp32 end-to-end (reference is fp32)
//   - GEMMs on the matrix pipe via V_WMMA_F32_16X16X4_F32 (wave32),
//     32x64 output tile per wave (2 A-frags x 4 B-frags -> 8 WMMA / K-step)
//   - edge aggregation: coalesced float4 gather + hardware
//     global_atomic_add_f32 (unsafeAtomicAdd); accumulators are L2-resident
//     (40MB/20MB << 192MB L2)
// Workspace layout (floats):
//   [0, 20480)                      inv_sqrt_deg  (N=20000, padded)
//   [20480, 20480+N*512)            h   (dense GEMM output, reused as h2)
//   [20480+N*512, 20480+2*N*512)    agg (post-aggregation layer-1 activations)
// Total ~82 MB.
// ---------------------------------------------------------------------------

#define GCN_N      20000
#define GCN_IN     256
#define GCN_HID    512
#define GCN_OUT    256

typedef __attribute__((ext_vector_type(2))) float v2f;
typedef __attribute__((ext_vector_type(8))) float v8f;

// ---------------- degree / normalization ----------------------------------

__global__ __launch_bounds__(256)
void k_deg_init(float* __restrict__ deg, int n) {
    int i = blockIdx.x * 256 + threadIdx.x;
    if (i < n) deg[i] = 1.0f;                      // self-loop contributes 1
}

__global__ __launch_bounds__(256)
void k_deg_edges(const int* __restrict__ dst, float* __restrict__ deg, int e) {
    int i = blockIdx.x * 256 + threadIdx.x;
    if (i < e) unsafeAtomicAdd(&deg[dst[i]], 1.0f);
}

__global__ __launch_bounds__(256)
void k_inv_sqrt(float* __restrict__ deg, int n) {
    int i = blockIdx.x * 256 + threadIdx.x;
    if (i < n) deg[i] = rsqrtf(fmaxf(deg[i], 1.0f));
}

// ---------------- fp32 WMMA GEMM: C[M,N] = A[M,K] * B[K,N] ----------------
// One wave computes a 32x64 strip: 2 M-subtiles x 4 N-subtiles of 16x16,
// sharing both A fragments (x4) and B fragments (x2).
// Requires M%32==0 (20000=625*32), K%4==0, N % (4 waves * 64) == 0.

__global__ __launch_bounds__(128)
void k_gemm_f32_wmma(const float* __restrict__ A, const float* __restrict__ B,
                     float* __restrict__ C, int M, int K, int N) {
    const int lane = threadIdx.x & 31;
    const int wave = threadIdx.x >> 5;      // 0..3
    const int l16  = lane & 15;
    const int hi   = lane >> 4;             // 0 or 1
    const int mBase = blockIdx.x * 32;
    const int nBase = (blockIdx.y * 4 + wave) * 64;
    if (mBase >= M || nBase >= N) return;   // wave-uniform; EXEC stays full

    // A fragment: lane l16 = row M, element j holds K = k0 + 2*hi + j
    const float* Ap0 = A + (size_t)(mBase + l16) * K + 2 * hi;
    const float* Ap1 = Ap0 + (size_t)16 * K;
    // B fragment: lane l16 = col N, element j holds K = k0 + 2*hi + j
    const float* Bp0 = B + (size_t)(2 * hi) * N + nBase + l16;

    v8f acc[2][4];
#pragma unroll
    for (int m = 0; m < 2; ++m)
#pragma unroll
        for (int t = 0; t < 4; ++t)
            acc[m][t] = (v8f){};

    for (int k0 = 0; k0 < K; k0 += 4) {
        v2f a0 = *(const v2f*)(Ap0 + k0);               // 8B aligned
        v2f a1 = *(const v2f*)(Ap1 + k0);
        const float* Bp = Bp0 + (size_t)k0 * N;
        v2f b[4];
#pragma unroll
        for (int t = 0; t < 4; ++t) {
            b[t][0] = Bp[t * 16];
            b[t][1] = Bp[N + t * 16];
        }
#pragma unroll
        for (int t = 0; t < 4; ++t) {
            acc[0][t] = __builtin_amdgcn_wmma_f32_16x16x4_f32(
                false, a0, false, b[t], (short)0, acc[0][t], false, false);
            acc[1][t] = __builtin_amdgcn_wmma_f32_16x16x4_f32(
                false, a1, false, b[t], (short)0, acc[1][t], false, false);
        }
    }
    // C/D layout: VGPR r, lanes 0-15 -> row r, lanes 16-31 -> row r+8
#pragma unroll
    for (int m = 0; m < 2; ++m) {
#pragma unroll
        for (int r = 0; r < 8; ++r) {
            float* Cp = C + (size_t)(mBase + m * 16 + r + 8 * hi) * N + nBase + l16;
            Cp[0]  = acc[m][0][r];
            Cp[16] = acc[m][1][r];
            Cp[32] = acc[m][2][r];
            Cp[48] = acc[m][3][r];
        }
    }
}

// ---------------- aggregation ---------------------------------------------

// out[i,f] = h[i,f] * inv[i]^2  (the self-loop message; also initializes out)
__global__ __launch_bounds__(256)
void k_selfloop_init(const float4* __restrict__ h, const float* __restrict__ inv,
                     float4* __restrict__ out, int total4, int f4) {
    int idx = blockIdx.x * 256 + threadIdx.x;
    if (idx >= total4) return;
    int i = idx / f4;
    float w = inv[i]; w *= w;
    float4 v = h[idx];
    v.x *= w; v.y *= w; v.z *= w; v.w *= w;
    out[idx] = v;
}

// one wave per edge: out[dst] += h[src] * inv[src]*inv[dst]
// float4 gather (global_load_b128, coalesced) + 4x f32 hardware atomics
__global__ __launch_bounds__(256)
void k_edge_scatter(const int* __restrict__ src, const int* __restrict__ dst,
                    const float* __restrict__ inv, const float* __restrict__ h,
                    float* __restrict__ out, int E, int F) {
    int e = blockIdx.x * 8 + (threadIdx.x >> 5);
    if (e >= E) return;
    int lane = threadIdx.x & 31;
    int s = src[e];
    int d = dst[e];
    float norm = inv[s] * inv[d];
    const float4* hs = (const float4*)(h + (size_t)s * F);
    float* od = out + (size_t)d * F;
    int f4 = F >> 2;
    for (int c = lane; c < f4; c += 32) {
        float4 v = hs[c];
        int base = c * 4;
        unsafeAtomicAdd(&od[base + 0], v.x * norm);
        unsafeAtomicAdd(&od[base + 1], v.y * norm);
        unsafeAtomicAdd(&od[base + 2], v.z * norm);
        unsafeAtomicAdd(&od[base + 3], v.w * norm);
    }
}

// y += bias (broadcast over rows), optional ReLU; float4 vectorized
__global__ __launch_bounds__(256)
void k_bias_act(float4* __restrict__ y, const float4* __restrict__ b,
                int total4, int f4, int do_relu) {
    int idx = blockIdx.x * 256 + threadIdx.x;
    if (idx >= total4) return;
    int c = idx % f4;
    float4 v = y[idx];
    float4 bb = b[c];
    v.x += bb.x; v.y += bb.y; v.z += bb.z; v.w += bb.w;
    if (do_relu) {
        v.x = fmaxf(v.x, 0.0f); v.y = fmaxf(v.y, 0.0f);
        v.z = fmaxf(v.z, 0.0f); v.w = fmaxf(v.w, 0.0f);
    }
    y[idx] = v;
}

// ---------------- launch ---------------------------------------------------

extern "C" void kernel_launch(void* const* d_in, const int* in_sizes, int n_in,
                              void* d_out, int out_size, void* d_ws, size_t ws_size,
                              hipStream_t stream) {
    const float* x  = (const float*)d_in[0];   // [N, 256]
    const int*   ei = (const int*)d_in[1];     // [2, E] (int per harness doc)
    const float* W1 = (const float*)d_in[2];   // [256, 512]
    const float* b1 = (const float*)d_in[3];   // [512]
    const float* W2 = (const float*)d_in[4];   // [512, 256]
    const float* b2 = (const float*)d_in[5];   // [256]
    const int E = in_sizes[1] / 2;
    const int* src = ei;
    const int* dst = ei + E;

    float* inv = (float*)d_ws;                         // N floats (padded 20480)
    float* h   = inv + 20480;                          // N*512
    float* agg = h + (size_t)GCN_N * GCN_HID;          // N*512
    float* out = (float*)d_out;                        // N*256

    // --- normalization coefficients ---
    k_deg_init <<<(GCN_N + 255) / 256, 256, 0, stream>>>(inv, GCN_N);
    k_deg_edges<<<(E + 255) / 256, 256, 0, stream>>>(dst, inv, E);
    k_inv_sqrt <<<(GCN_N + 255) / 256, 256, 0, stream>>>(inv, GCN_N);

    // --- layer 1: h = x @ W1 ; agg = scatter(norm * h) ; relu(agg + b1) ---
    {
        dim3 grid(GCN_N / 32, GCN_HID / 256);
        k_gemm_f32_wmma<<<grid, 128, 0, stream>>>(x, W1, h, GCN_N, GCN_IN, GCN_HID);
        int total4 = GCN_N * (GCN_HID / 4);
        k_selfloop_init<<<(total4 + 255) / 256, 256, 0, stream>>>(
            (const float4*)h, inv, (float4*)agg, total4, GCN_HID / 4);
        k_edge_scatter<<<(E + 7) / 8, 256, 0, stream>>>(src, dst, inv, h, agg, E, GCN_HID);
        k_bias_act<<<(total4 + 255) / 256, 256, 0, stream>>>(
            (float4*)agg, (const float4*)b1, total4, GCN_HID / 4, 1);
    }

    // --- layer 2: h2 = agg @ W2 ; out = scatter(norm * h2) + b2 ---
    {
        float* h2 = h;  // reuse: only N*256 of the N*512 region
        dim3 grid(GCN_N / 32, GCN_OUT / 256);
        k_gemm_f32_wmma<<<grid, 128, 0, stream>>>(agg, W2, h2, GCN_N, GCN_HID, GCN_OUT);
        int total4 = GCN_N * (GCN_OUT / 4);
        k_selfloop_init<<<(total4 + 255) / 256, 256, 0, stream>>>(
            (const float4*)h2, inv, (float4*)out, total4, GCN_OUT / 4);
        k_edge_scatter<<<(E + 7) / 8, 256, 0, stream>>>(src, dst, inv, h2, out, E, GCN_OUT);
        k_bias_act<<<(total4 + 255) / 256, 256, 0, stream>>>(
            (float4*)out, (const float4*)b2, total4, GCN_OUT / 4, 0);
    }
}